// GCN_54150947668273
// MI455X (gfx1250) — compile-verified
//
#include <hip/hip_runtime.h>
#include <hip/hip_bf16.h>

#define N_NODES   100000
#define N_EDGES   1600000
#define HID       128
#define N_CLASSES 10
#define N_GRAPHS  512
#define N_LAYERS  5
#define E_TOT     (N_EDGES + N_NODES)   // edges + self loops

typedef __attribute__((ext_vector_type(2))) float v2f;
typedef __attribute__((ext_vector_type(8))) float v8f;

// ---------------------------------------------------------------------------
// Degree / normalization
// ---------------------------------------------------------------------------
__global__ void init_deg_kernel(float* __restrict__ deg) {
    int i = blockIdx.x * blockDim.x + threadIdx.x;
    if (i < N_NODES) deg[i] = 1.0f;   // self-loop contributes 1
}

__global__ void deg_accum_kernel(const int* __restrict__ dst, float* __restrict__ deg) {
    int e = blockIdx.x * blockDim.x + threadIdx.x;
    if (e < N_EDGES) atomicAdd(&deg[dst[e]], 1.0f);
}

__global__ void dinv_kernel(float* __restrict__ deg) {
    int i = blockIdx.x * blockDim.x + threadIdx.x;
    if (i < N_NODES) deg[i] = rsqrtf(deg[i]);   // deg >= 1 always
}

__global__ void norm_kernel(const int* __restrict__ src, const int* __restrict__ dst,
                            const float* __restrict__ dinv, float* __restrict__ norm) {
    int e = blockIdx.x * blockDim.x + threadIdx.x;
    if (e >= E_TOT) return;
    if (e < N_EDGES) {
        norm[e] = dinv[src[e]] * dinv[dst[e]];
    } else {
        float dv = dinv[e - N_EDGES];
        norm[e] = dv * dv;
    }
}

// ---------------------------------------------------------------------------
// FP32 WMMA GEMM: H[N x 128] = X[N x 128] @ W[128 x 128]
// One wave computes one 16x16 output tile; K swept in steps of 4 via
// V_WMMA_F32_16X16X4_F32 (32 chained WMMAs per tile).
// A 16x4 layout (ISA 7.12.2): lane m=lane&15 -> row, kh=lane>>4 -> k pair,
//   vgpr j holds K = kbase + 2*kh + j  => one float2 load per lane.
// B 4x16 mirrored: lane&15 -> column N, kh/vgpr -> K.
// C/D 16x16: vgpr r holds row r + 8*kh, column lane&15.
// ---------------------------------------------------------------------------
__global__ __launch_bounds__(256)
void gemm_wmma_kernel(const float* __restrict__ X, const float* __restrict__ W,
                      float* __restrict__ H) {
    int wave     = (blockIdx.x * blockDim.x + threadIdx.x) >> 5;
    int lane     = threadIdx.x & 31;
    int row_tile = wave >> 3;            // HID/16 = 8 column tiles per row tile
    int col_tile = wave & 7;
    if (row_tile * 16 >= N_NODES) return;   // wave-uniform; EXEC stays all-ones

    int m  = lane & 15;
    int kh = lane >> 4;
    const float* xrow = X + (size_t)(row_tile * 16 + m) * HID;
    int cb = col_tile * 16;

    v8f acc = {};
    #pragma unroll
    for (int kb = 0; kb < HID; kb += 4) {
        int k0 = kb + 2 * kh;
        v2f a = *(const v2f*)(xrow + k0);
        v2f b;
        b.x = W[(size_t)(k0 + 0) * HID + cb + m];
        b.y = W[(size_t)(k0 + 1) * HID + cb + m];
        acc = __builtin_amdgcn_wmma_f32_16x16x4_f32(
            /*neg_a=*/false, a, /*neg_b=*/false, b,
            /*c_mod=*/(short)0, acc, /*reuse_a=*/false, /*reuse_b=*/false);
    }

    float* hbase = H + (size_t)(row_tile * 16 + 8 * kh) * HID + cb + m;
    #pragma unroll
    for (int r = 0; r < 8; ++r) hbase[(size_t)r * HID] = acc[r];
}

// ---------------------------------------------------------------------------
// Zero fill (float4 vectorized)
// ---------------------------------------------------------------------------
__global__ void zero_kernel(float4* __restrict__ p, int n4) {
    int i = blockIdx.x * blockDim.x + threadIdx.x;
    if (i < n4) p[i] = make_float4(0.f, 0.f, 0.f, 0.f);
}

// ---------------------------------------------------------------------------
// Edge scatter: out[dst] += h[src] * norm.  One wave per edge, lane owns a
// float4 (32 lanes * 4 = 128 feats).  Edges >= N_EDGES are self loops.
// ---------------------------------------------------------------------------
__global__ __launch_bounds__(256)
void scatter_kernel(const float* __restrict__ H, const int* __restrict__ src,
                    const int* __restrict__ dst, const float* __restrict__ norm,
                    float* __restrict__ out) {
    int wave = (blockIdx.x * blockDim.x + threadIdx.x) >> 5;
    int lane = threadIdx.x & 31;
    if (wave >= E_TOT) return;
    int s, d;
    if (wave < N_EDGES) { s = src[wave]; d = dst[wave]; }
    else                { s = wave - N_EDGES; d = s; }
    float w = norm[wave];
    float4 v = ((const float4*)(H + (size_t)s * HID))[lane];
    float* op = out + (size_t)d * HID + lane * 4;
    atomicAdd(op + 0, v.x * w);
    atomicAdd(op + 1, v.y * w);
    atomicAdd(op + 2, v.z * w);
    atomicAdd(op + 3, v.w * w);
}

// ---------------------------------------------------------------------------
// Fused bias + ReLU (in place)
// ---------------------------------------------------------------------------
__global__ void bias_relu_kernel(float* __restrict__ X, const float* __restrict__ b) {
    int i = blockIdx.x * blockDim.x + threadIdx.x;
    if (i < N_NODES * HID) {
        float v = X[i] + b[i & (HID - 1)];
        X[i] = v > 0.f ? v : 0.f;
    }
}

// ---------------------------------------------------------------------------
// Mean-pool accumulate: one wave per node, float4 per lane + count atomic
// ---------------------------------------------------------------------------
__global__ __launch_bounds__(256)
void pool_accum_kernel(const float* __restrict__ X, const int* __restrict__ batch,
                       float* __restrict__ pooled, float* __restrict__ counts) {
    int wave = (blockIdx.x * blockDim.x + threadIdx.x) >> 5;
    int lane = threadIdx.x & 31;
    if (wave >= N_NODES) return;
    int g = batch[wave];
    float4 v = ((const float4*)(X + (size_t)wave * HID))[lane];
    float* pp = pooled + (size_t)g * HID + lane * 4;
    atomicAdd(pp + 0, v.x);
    atomicAdd(pp + 1, v.y);
    atomicAdd(pp + 2, v.z);
    atomicAdd(pp + 3, v.w);
    if (lane == 0) atomicAdd(&counts[g], 1.0f);
}

// ---------------------------------------------------------------------------
// Final linear: out[g][c] = (pooled[g]/max(cnt,1)) . linW[:,c] + linb[c]
// ---------------------------------------------------------------------------
__global__ void final_linear_kernel(const float* __restrict__ pooled,
                                    const float* __restrict__ counts,
                                    const float* __restrict__ linW,
                                    const float* __restrict__ linb,
                                    float* __restrict__ out) {
    int i = blockIdx.x * blockDim.x + threadIdx.x;
    if (i >= N_GRAPHS * N_CLASSES) return;
    int g = i / N_CLASSES;
    int c = i - g * N_CLASSES;
    float scale = 1.0f / fmaxf(counts[g], 1.0f);
    const float* pg = pooled + (size_t)g * HID;
    float acc = 0.f;
    #pragma unroll 4
    for (int f = 0; f < HID; ++f) acc += pg[f] * linW[f * N_CLASSES + c];
    out[i] = acc * scale + linb[c];
}

// ---------------------------------------------------------------------------
// Host orchestration
// ---------------------------------------------------------------------------
extern "C" void kernel_launch(void* const* d_in, const int* in_sizes, int n_in,
                              void* d_out, int out_size, void* d_ws, size_t ws_size,
                              hipStream_t stream) {
    const float* x     = (const float*)d_in[0];
    const int*   edge  = (const int*)d_in[1];
    const int*   src   = edge;
    const int*   dst   = edge + N_EDGES;
    const int*   batch = (const int*)d_in[2];
    const float* Ws    = (const float*)d_in[3];
    const float* bs    = (const float*)d_in[4];
    const float* linW  = (const float*)d_in[5];
    const float* linb  = (const float*)d_in[6];
    float*       out   = (float*)d_out;

    // Workspace layout (floats); all offsets 16-byte aligned.
    float* ws = (float*)d_ws;
    float* deg    = ws;                                   // N_NODES
    float* norm   = deg + N_NODES;                        // E_TOT
    float* H      = norm + E_TOT;                         // N_NODES*HID
    float* X1     = H + (size_t)N_NODES * HID;            // N_NODES*HID
    float* X2     = X1 + (size_t)N_NODES * HID;           // N_NODES*HID
    float* pooled = X2 + (size_t)N_NODES * HID;           // N_GRAPHS*HID
    float* counts = pooled + (size_t)N_GRAPHS * HID;      // N_GRAPHS

    const int T = 256;

    // --- degree / normalization (computed once, reused all layers) ---
    init_deg_kernel<<<(N_NODES + T - 1) / T, T, 0, stream>>>(deg);
    deg_accum_kernel<<<(N_EDGES + T - 1) / T, T, 0, stream>>>(dst, deg);
    dinv_kernel<<<(N_NODES + T - 1) / T, T, 0, stream>>>(deg);
    norm_kernel<<<(E_TOT + T - 1) / T, T, 0, stream>>>(src, dst, deg, norm);

    // --- GCN layers ---
    const int gemm_blocks    = (N_NODES / 16) * (HID / 16) / 8;  // waves/8
    const int scatter_blocks = E_TOT / 8;                        // 1 wave/edge
    const int feat4          = N_NODES * HID / 4;

    const float* xin = x;
    float* xbuf[2] = { X1, X2 };
    for (int l = 0; l < N_LAYERS; ++l) {
        float* xout = xbuf[l & 1];
        const float* Wl = Ws + (size_t)l * HID * HID;
        const float* bl = bs + (size_t)l * HID;

        gemm_wmma_kernel<<<gemm_blocks, T, 0, stream>>>(xin, Wl, H);
        zero_kernel<<<(feat4 + T - 1) / T, T, 0, stream>>>((float4*)xout, feat4);
        scatter_kernel<<<scatter_blocks, T, 0, stream>>>(H, src, dst, norm, xout);
        bias_relu_kernel<<<(N_NODES * HID + T - 1) / T, T, 0, stream>>>(xout, bl);

        xin = xout;
    }

    // --- global mean pool + classifier ---
    int pool4 = (N_GRAPHS * HID + N_GRAPHS + 3) / 4;   // pooled + counts (contiguous)
    zero_kernel<<<(pool4 + T - 1) / T, T, 0, stream>>>((float4*)pooled, pool4);
    pool_accum_kernel<<<N_NODES / 8, T, 0, stream>>>(xin, batch, pooled, counts);
    final_linear_kernel<<<(N_GRAPHS * N_CLASSES + T - 1) / T, T, 0, stream>>>(
        pooled, counts, linW, linb, out);
}